// SS1D_76106820485661
// MI455X (gfx1250) — compile-verified
//
#include <hip/hip_runtime.h>
#include <hip/hip_bf16.h>
#include <math.h>

// ---------------- problem constants (match reference) ----------------
#define BSZ   4
#define LSEQ  4096
#define DMOD  96
#define DIN   192      // d_inner
#define NST   64       // d_state
#define DTRK  6        // dt_rank
#define KDIR  2
#define CROW  134      // dt_rank + 2*d_state

// ---------------- fp32 WMMA helpers (V_WMMA_F32_16X16X4_F32) ----------------
typedef float v2f __attribute__((ext_vector_type(2)));
typedef float v8f __attribute__((ext_vector_type(8)));

// Layouts per CDNA5 ISA 7.12.2:
//  A (16x4, MxK): lanes 0-15 hold M=lane, VGPR0=K(0|2), VGPR1=K(1|3) (hi half-lanes K+2)
//  B (4x16, KxN): mirrored — lanes 0-15 hold N=lane, VGPR0/1 = K pair per half
//  C/D (16x16):   lanes 0-15 N=lane (hi: N=lane-16, M+=8); VGPR r -> M=r
__device__ __forceinline__ v8f wmma4(v2f a, v2f b, v8f c) {
  return __builtin_amdgcn_wmma_f32_16x16x4_f32(false, a, false, b, (short)0, c, false, false);
}

__device__ __forceinline__ float silu_f(float v) { return v / (1.f + __expf(-v)); }

// ---------------- CDNA5 async copy (ASYNCcnt) helpers ----------------
// Generic pointer to an LDS object: low 32 bits are the wave-relative LDS byte
// address (ISA 10.2).
__device__ __forceinline__ uint32_t lds_off32(const void* p) {
  return (uint32_t)(uintptr_t)p;
}
// GLOBAL_LOAD_ASYNC_TO_LDS_B128: VDST = LDS byte address VGPR, VADDR = 64-bit
// global address, SADDR disabled ("off"). Tracked with ASYNCcnt.
__device__ __forceinline__ void async_ld_b128(uint32_t lds, const void* g) {
  asm volatile("global_load_async_to_lds_b128 %0, %1, off"
               :: "v"(lds), "v"((unsigned long long)(uintptr_t)g)
               : "memory");
}

// =====================================================================
// K1: in_proj  C(16384,384) = x(16384,96) @ W_in^T ; split xi / silu(z)
// =====================================================================
__global__ __launch_bounds__(256) void k_inproj(const float* __restrict__ x,
                                                const float* __restrict__ Win,
                                                float* __restrict__ xi,
                                                float* __restrict__ zs) {
  const int lane = threadIdx.x & 31;
  const int wg   = blockIdx.x * 8 + (threadIdx.x >> 5);
  const int ntile = wg % 24, mtile = wg / 24;          // N=384 -> 24 tiles, M=16384 -> 1024
  const int m0 = mtile * 16, n0 = ntile * 16;
  const int row = m0 + (lane & 15);
  const int col = n0 + (lane & 15);
  v8f acc = {};
  #pragma unroll
  for (int k0 = 0; k0 < DMOD; k0 += 4) {
    const int kk = k0 + ((lane >> 4) << 1);
    v2f a, b;
    a.x = x[row * DMOD + kk];     a.y = x[row * DMOD + kk + 1];
    b.x = Win[col * DMOD + kk];   b.y = Win[col * DMOD + kk + 1];
    acc = wmma4(a, b, acc);
  }
  const int n  = n0 + (lane & 15);
  const int mb = m0 + ((lane >> 4) << 3);
  if (n0 < DIN) {
    #pragma unroll
    for (int r = 0; r < 8; ++r) xi[(mb + r) * DIN + n] = acc[r];
  } else {
    const int nz = n - DIN;
    #pragma unroll
    for (int r = 0; r < 8; ++r) zs[(mb + r) * DIN + nz] = silu_f(acc[r]);
  }
}

// =====================================================================
// K2: depthwise conv3 (+bias,+silu) with LDS transpose (B,L,D)->(B,D,L)
// =====================================================================
__global__ __launch_bounds__(256) void k_conv(const float* __restrict__ xi,
                                              const float* __restrict__ cw,
                                              const float* __restrict__ cb,
                                              float* __restrict__ xc) {
  __shared__ float tile[34 * 193];                      // stride 193 => bank-conflict free
  const int b = blockIdx.y, l0 = blockIdx.x * 32;
  for (int idx = threadIdx.x; idx < 34 * DIN; idx += 256) {
    const int r = idx / DIN, d = idx - r * DIN;
    const int l = l0 + r - 1;
    tile[r * 193 + d] = (l >= 0 && l < LSEQ) ? xi[(b * LSEQ + l) * DIN + d] : 0.f;
  }
  __syncthreads();
  for (int idx = threadIdx.x; idx < 32 * DIN; idx += 256) {
    const int lc = idx & 31, d = idx >> 5;
    float v = cw[d * 3 + 0] * tile[(lc    ) * 193 + d]
            + cw[d * 3 + 1] * tile[(lc + 1) * 193 + d]
            + cw[d * 3 + 2] * tile[(lc + 2) * 193 + d] + cb[d];
    xc[(b * DIN + d) * LSEQ + l0 + lc] = silu_f(v);
  }
}

// =====================================================================
// K3: x_proj  (134,192) @ xc_b(192,L) per (b,k); outputs stored in SCAN
// order (k=1 columns flipped): dt_raw (B,K,6,L), B^T/C^T (B,K,L,64)
// =====================================================================
__global__ __launch_bounds__(256) void k_xproj(const float* __restrict__ xpw,
                                               const float* __restrict__ xc,
                                               float* __restrict__ dtr,
                                               float* __restrict__ bsT,
                                               float* __restrict__ csT) {
  const int lane = threadIdx.x & 31;
  int wg = blockIdx.x * 8 + (threadIdx.x >> 5);
  const int ntile = wg % 256;  wg /= 256;               // L/16 = 256
  const int mtile = wg % 9;    wg /= 9;                 // ceil(134/16) = 9
  const int k = wg & 1, b = wg >> 1;
  const int m0 = mtile * 16, l0 = ntile * 16;
  const float* A  = xpw + k * CROW * DIN;
  const float* Bm = xc + b * DIN * LSEQ;
  const int row = m0 + (lane & 15);
  const int col = l0 + (lane & 15);
  const bool rok = row < CROW;
  v8f acc = {};
  #pragma unroll 8
  for (int k0 = 0; k0 < DIN; k0 += 4) {
    const int kk = k0 + ((lane >> 4) << 1);
    v2f a, bb;
    a.x = rok ? A[row * DIN + kk]     : 0.f;
    a.y = rok ? A[row * DIN + kk + 1] : 0.f;
    bb.x = Bm[kk * LSEQ + col];
    bb.y = Bm[(kk + 1) * LSEQ + col];
    acc = wmma4(a, bb, acc);
  }
  const int n    = l0 + (lane & 15);
  const int mb   = m0 + ((lane >> 4) << 3);
  const int lpos = (k == 0) ? n : (LSEQ - 1 - n);       // scan-order column
  const int bk   = b * KDIR + k;
  #pragma unroll
  for (int r = 0; r < 8; ++r) {
    const int c = mb + r;
    if (c >= CROW) continue;
    const float v = acc[r];
    if (c < DTRK)            dtr[(bk * DTRK + c) * LSEQ + lpos] = v;
    else if (c < DTRK + NST) bsT[(bk * LSEQ + lpos) * NST + (c - DTRK)] = v;
    else                     csT[(bk * LSEQ + lpos) * NST + (c - DTRK - NST)] = v;
  }
}

// =====================================================================
// K4: dt proj  delta(192,L) = dt_w(192,6) @ dt_raw(6,L)  (K padded 6->8)
// =====================================================================
__global__ __launch_bounds__(256) void k_dtproj(const float* __restrict__ dpw,
                                                const float* __restrict__ dtr,
                                                float* __restrict__ delta) {
  const int lane = threadIdx.x & 31;
  int wg = blockIdx.x * 8 + (threadIdx.x >> 5);
  const int ntile = wg % 256;  wg /= 256;
  const int mtile = wg % 12;   wg /= 12;                // 192/16
  const int k = wg & 1, b = wg >> 1;
  const int m0 = mtile * 16, l0 = ntile * 16;
  const float* A  = dpw + k * DIN * DTRK;
  const float* Bm = dtr + (b * KDIR + k) * DTRK * LSEQ;
  const int row = m0 + (lane & 15);
  const int col = l0 + (lane & 15);
  v8f acc = {};
  { // K = 0..3
    const int kk = (lane >> 4) << 1;
    v2f a, bb;
    a.x = A[row * DTRK + kk];        a.y = A[row * DTRK + kk + 1];
    bb.x = Bm[kk * LSEQ + col];      bb.y = Bm[(kk + 1) * LSEQ + col];
    acc = wmma4(a, bb, acc);
  }
  { // K = 4..7 (K=6,7 zero-padded)
    v2f a, bb;
    if (lane < 16) {
      a.x = A[row * DTRK + 4];  a.y = A[row * DTRK + 5];
      bb.x = Bm[4 * LSEQ + col]; bb.y = Bm[5 * LSEQ + col];
    } else { a.x = 0.f; a.y = 0.f; bb.x = 0.f; bb.y = 0.f; }
    acc = wmma4(a, bb, acc);
  }
  const int n  = l0 + (lane & 15);
  const int mb = m0 + ((lane >> 4) << 3);
  #pragma unroll
  for (int r = 0; r < 8; ++r)
    delta[((b * KDIR + k) * DIN + (mb + r)) * LSEQ + n] = acc[r];
}

// =====================================================================
// K5: selective scan. One wave32 per (b,k,d); lane owns states n, n+32.
// 16 channels/block share 32-step B/C chunks staged in LDS with
// double-buffered GLOBAL_LOAD_ASYNC_TO_LDS_B128 (ASYNCcnt).
// =====================================================================
#define CHUNK_F (32 * NST)      // floats per chunk region (8 KB)

__global__ __launch_bounds__(512) void k_scan(const float* __restrict__ xc,
                                              const float* __restrict__ delta,
                                              const float* __restrict__ bsT,
                                              const float* __restrict__ csT,
                                              const float* __restrict__ A_logs,
                                              const float* __restrict__ Dsv,
                                              const float* __restrict__ dt_bias,
                                              float* __restrict__ ys) {
  __shared__ __align__(16) float Bs_s[2][CHUNK_F];
  __shared__ __align__(16) float Cs_s[2][CHUNK_F];
  __shared__ float ysm[16 * 32];
  const int tid  = threadIdx.x;
  const int lane = tid & 31;
  const int w    = tid >> 5;
  const int k = blockIdx.y, b = blockIdx.z;
  const int d = blockIdx.x * 16 + w;
  const int kd = k * DIN + d;
  const float A0 = -__expf(A_logs[kd * NST + lane]);
  const float A1 = -__expf(A_logs[kd * NST + lane + 32]);
  const float Dskip = Dsv[kd];
  const float bias  = dt_bias[kd];
  const float* up = xc    + (b * DIN + d) * LSEQ;                 // unflipped u
  const float* dp = delta + ((b * KDIR + k) * DIN + d) * LSEQ;    // scan order
  const float* bp = bsT   + (b * KDIR + k) * LSEQ * NST;
  const float* cp = csT   + (b * KDIR + k) * LSEQ * NST;
  float* yp       = ys    + ((b * KDIR + k) * DIN + d) * LSEQ;
  float h0 = 0.f, h1 = 0.f;

  // Each of the 512 threads async-copies one 16B slice of each 8KB region:
  // 2 ASYNC ops outstanding per wave per in-flight chunk.
  const uint32_t ldsB0 = lds_off32(&Bs_s[0][tid * 4]);
  const uint32_t ldsB1 = lds_off32(&Bs_s[1][tid * 4]);
  const uint32_t ldsC0 = lds_off32(&Cs_s[0][tid * 4]);
  const uint32_t ldsC1 = lds_off32(&Cs_s[1][tid * 4]);

  // prologue: chunk 0 -> buffer 0
  async_ld_b128(ldsB0, bp + tid * 4);
  async_ld_b128(ldsC0, cp + tid * 4);

  const int nc = LSEQ / 32;
  for (int c = 0; c < nc; ++c) {
    const int buf = c & 1;
    // prefetch chunk c+1 into the other buffer while we compute chunk c
    if (c + 1 < nc) {
      const int off = (c + 1) * CHUNK_F + tid * 4;
      async_ld_b128(buf ? ldsB0 : ldsB1, bp + off);
      async_ld_b128(buf ? ldsC0 : ldsC1, cp + off);
    }
    // per-wave u/delta registers for this chunk (regular loads)
    const int ls = c * 32 + lane;
    const float u_c  = (k == 0) ? up[ls] : up[LSEQ - 1 - ls];
    const float dt_c = dp[ls];
    // async loads complete in order: <=2 outstanding => chunk c has landed
    if (c + 1 < nc) asm volatile("s_wait_asynccnt 0x2" ::: "memory");
    else            asm volatile("s_wait_asynccnt 0x0" ::: "memory");
    __syncthreads();
    const float* Bsh = Bs_s[buf];
    const float* Csh = Cs_s[buf];
    #pragma unroll 4
    for (int t = 0; t < 32; ++t) {
      const float ut  = __shfl(u_c,  t, 32);
      float dtv = __shfl(dt_c, t, 32) + bias;
      dtv = (dtv > 20.f) ? dtv : log1pf(__expf(dtv));             // softplus
      const float bs0 = Bsh[t * NST + lane], bs1 = Bsh[t * NST + 32 + lane];
      const float cs0 = Csh[t * NST + lane], cs1 = Csh[t * NST + 32 + lane];
      const float du = dtv * ut;
      h0 = __expf(dtv * A0) * h0 + du * bs0;
      h1 = __expf(dtv * A1) * h1 + du * bs1;
      float p = h0 * cs0 + h1 * cs1;
      #pragma unroll
      for (int o = 16; o > 0; o >>= 1) p += __shfl_xor(p, o, 32); // y = h . C
      p += Dskip * ut;
      if (lane == t) ysm[w * 32 + t] = p;                         // self write/read per lane
    }
    yp[c * 32 + lane] = ysm[w * 32 + lane];
    __syncthreads();   // everyone done reading buf before it is overwritten
  }
}

// =====================================================================
// K6: merge (fwd + flipped bwd) + LayerNorm(192) + z-gate -> yg(B,L,192)
// =====================================================================
__global__ __launch_bounds__(256) void k_merge_ln(const float* __restrict__ ys,
                                                  const float* __restrict__ zs,
                                                  const float* __restrict__ ln_g,
                                                  const float* __restrict__ ln_b,
                                                  float* __restrict__ yg) {
  __shared__ float  yt[16 * 193];
  __shared__ float2 red[256];
  __shared__ float2 mv[16];
  const int b = blockIdx.y, l0 = blockIdx.x * 16;
  const float* y0 = ys + (b * KDIR + 0) * DIN * LSEQ;
  const float* y1 = ys + (b * KDIR + 1) * DIN * LSEQ;
  for (int idx = threadIdx.x; idx < 16 * DIN; idx += 256) {
    const int lt = idx & 15, d = idx >> 4, l = l0 + lt;
    yt[lt * 193 + d] = y0[d * LSEQ + l] + y1[d * LSEQ + (LSEQ - 1 - l)];
  }
  __syncthreads();
  {
    const int row = threadIdx.x >> 4, cix = threadIdx.x & 15;
    float s = 0.f, q = 0.f;
    for (int d = cix; d < DIN; d += 16) { const float v = yt[row * 193 + d]; s += v; q += v * v; }
    red[threadIdx.x] = make_float2(s, q);
  }
  __syncthreads();
  if ((threadIdx.x & 15) == 0) {
    const int row = threadIdx.x >> 4;
    float s = 0.f, q = 0.f;
    for (int i = 0; i < 16; ++i) { const float2 r = red[row * 16 + i]; s += r.x; q += r.y; }
    const float mu  = s * (1.f / DIN);
    const float var = q * (1.f / DIN) - mu * mu;
    mv[row] = make_float2(mu, rsqrtf(var + 1e-5f));
  }
  __syncthreads();
  for (int idx = threadIdx.x; idx < 16 * DIN; idx += 256) {
    const int d = idx % DIN, row = idx / DIN;
    const float2 m = mv[row];
    float v = (yt[row * 193 + d] - m.x) * m.y * ln_g[d] + ln_b[d];
    v *= zs[(b * LSEQ + l0 + row) * DIN + d];
    yg[(b * LSEQ + l0 + row) * DIN + d] = v;
  }
}

// =====================================================================
// K7: out_proj  out(16384,96) = yg(16384,192) @ W_out^T
// =====================================================================
__global__ __launch_bounds__(256) void k_outproj(const float* __restrict__ yg,
                                                 const float* __restrict__ Wout,
                                                 float* __restrict__ out) {
  const int lane = threadIdx.x & 31;
  const int wg = blockIdx.x * 8 + (threadIdx.x >> 5);
  const int ntile = wg % 6, mtile = wg / 6;             // N=96 -> 6 tiles
  const int m0 = mtile * 16, n0 = ntile * 16;
  const int row = m0 + (lane & 15);
  const int col = n0 + (lane & 15);
  v8f acc = {};
  #pragma unroll 8
  for (int k0 = 0; k0 < DIN; k0 += 4) {
    const int kk = k0 + ((lane >> 4) << 1);
    v2f a, b;
    a.x = yg[row * DIN + kk];    a.y = yg[row * DIN + kk + 1];
    b.x = Wout[col * DIN + kk];  b.y = Wout[col * DIN + kk + 1];
    acc = wmma4(a, b, acc);
  }
  const int n  = n0 + (lane & 15);
  const int mb = m0 + ((lane >> 4) << 3);
  #pragma unroll
  for (int r = 0; r < 8; ++r) out[(mb + r) * DMOD + n] = acc[r];
}

// =====================================================================
extern "C" void kernel_launch(void* const* d_in, const int* in_sizes, int n_in,
                              void* d_out, int out_size, void* d_ws, size_t ws_size,
                              hipStream_t stream) {
  (void)in_sizes; (void)n_in; (void)out_size; (void)ws_size;
  const float* x     = (const float*)d_in[0];
  const float* Win   = (const float*)d_in[1];
  const float* cw    = (const float*)d_in[2];
  const float* cb    = (const float*)d_in[3];
  const float* xpw   = (const float*)d_in[4];
  const float* dpw   = (const float*)d_in[5];
  const float* dpb   = (const float*)d_in[6];
  const float* Alogs = (const float*)d_in[7];
  const float* Dsv   = (const float*)d_in[8];
  const float* lng   = (const float*)d_in[9];
  const float* lnb   = (const float*)d_in[10];
  const float* Wout  = (const float*)d_in[11];
  float* out = (float*)d_out;
  float* ws  = (float*)d_ws;

  // workspace layout (floats); total ~26.4M floats = ~106 MB
  const size_t szBLD = (size_t)BSZ * LSEQ * DIN;          // 3,145,728
  float* xi    = ws;                                      // (B,L,192)
  float* zs    = xi  + szBLD;                             // (B,L,192) silu(z)
  float* xc    = zs  + szBLD;                             // (B,192,L)
  float* dtr   = xc  + szBLD;                             // (B,K,6,L)
  float* bsT   = dtr + (size_t)BSZ * KDIR * DTRK * LSEQ;  // (B,K,L,64)
  float* csT   = bsT + (size_t)BSZ * KDIR * LSEQ * NST;   // (B,K,L,64)
  float* delta = csT + (size_t)BSZ * KDIR * LSEQ * NST;   // (B,K,192,L)
  float* ysbuf = delta + (size_t)BSZ * KDIR * DIN * LSEQ; // (B,K,192,L)
  float* yg    = xi;                                      // alias: xi dead after conv

  k_inproj  <<<3072, 256, 0, stream>>>(x, Win, xi, zs);
  k_conv    <<<dim3(LSEQ / 32, BSZ), 256, 0, stream>>>(xi, cw, cb, xc);
  k_xproj   <<<2304, 256, 0, stream>>>(xpw, xc, dtr, bsT, csT);
  k_dtproj  <<<3072, 256, 0, stream>>>(dpw, dtr, delta);
  k_scan    <<<dim3(DIN / 16, KDIR, BSZ), 512, 0, stream>>>(xc, delta, bsT, csT,
                                                            Alogs, Dsv, dpb, ysbuf);
  k_merge_ln<<<dim3(LSEQ / 16, BSZ), 256, 0, stream>>>(ysbuf, zs, lng, lnb, yg);
  k_outproj <<<768, 256, 0, stream>>>(yg, Wout, out);
}